// GCN_55018531062470
// MI455X (gfx1250) — compile-verified
//
#include <hip/hip_runtime.h>
#include <hip/hip_bf16.h>

#define N_NODES 100000
#define N_EDGES 1600000
#define IN_CH   128
#define HID_CH  64
#define OUT_CH  32

typedef __attribute__((ext_vector_type(2))) float f32x2;
typedef __attribute__((ext_vector_type(8))) float f32x8;

__device__ __forceinline__ void atomAddF(float* p, float v) {
    // lowers to global_atomic_add_f32 (no CAS loop)
    unsafeAtomicAdd(p, v);
}

// ---------------- degree / normalization ----------------

__global__ void deg_init_kernel(float* deg, int n) {
    int i = blockIdx.x * blockDim.x + threadIdx.x;
    if (i < n) deg[i] = 1.0f;   // self-loop
}

__global__ void deg_count_kernel(const int* __restrict__ dst, float* deg, int nE) {
    int e = blockIdx.x * blockDim.x + threadIdx.x;
    if (e < nE) atomAddF(&deg[dst[e]], 1.0f);
}

__global__ void rsqrt_kernel(float* deg, int n) {
    int i = blockIdx.x * blockDim.x + threadIdx.x;
    if (i < n) deg[i] = rsqrtf(deg[i]);
}

// ---------------- fp32 WMMA GEMM: C[rows x NCOL] = A[rows x K] * B[K x NCOL] ----------------
// One block handles 16 rows; wave w handles column tile [w*16, w*16+16).
// blockDim.x = (NCOL/16) * 32. gridDim.x = rows/16 (exact).

template <int K, int NCOL>
__global__ void gemm_wmma_f32(const float* __restrict__ A,
                              const float* __restrict__ B,
                              float* __restrict__ C) {
    const int wave = threadIdx.x >> 5;
    const int lane = threadIdx.x & 31;
    const int row0 = blockIdx.x << 4;

    // A 16x4 fp32 layout: lanes 0-15 -> K={k0,k0+1}, lanes 16-31 -> K={k0+2,k0+3}; M = lane&15
    const int m  = lane & 15;
    const int kk = (lane >> 4) << 1;          // 0 or 2
    const int n  = (lane & 15) + (wave << 4); // output column

    const float* ap = A + (size_t)(row0 + m) * K + kk;
    const float* bp = B + (size_t)kk * NCOL + n;

    f32x8 acc = {0.f, 0.f, 0.f, 0.f, 0.f, 0.f, 0.f, 0.f};

    for (int k0 = 0; k0 < K; k0 += 4) {
        f32x2 a, b;
        a.x = ap[k0];
        a.y = ap[k0 + 1];
        b.x = bp[(size_t)k0 * NCOL];
        b.y = bp[(size_t)(k0 + 1) * NCOL];
        acc = __builtin_amdgcn_wmma_f32_16x16x4_f32(
            /*neg_a=*/false, a, /*neg_b=*/false, b,
            /*c_mod=*/(short)0, acc, /*reuse_a=*/false, /*reuse_b=*/false);
    }

    // C/D layout: VGPR i -> M = i (lanes 0-15) or 8+i (lanes 16-31); N = lane&15 (+ wave tile)
    const int mbase = (lane >> 4) << 3; // 0 or 8
    float* cp = C + (size_t)row0 * NCOL + n;
#pragma unroll
    for (int i = 0; i < 8; ++i)
        cp[(size_t)(mbase + i) * NCOL] = acc[i];
}

// ---------------- edge scatter: agg[dst] += h[src] * dinv[src]*dinv[dst] ----------------
// C consecutive threads per edge (C = 64 or 32, power of two; grid exact).

template <int C, int LOG2C>
__global__ void edge_agg_kernel(const float* __restrict__ h,
                                const int* __restrict__ src,
                                const int* __restrict__ dst,
                                const float* __restrict__ dinv,
                                float* agg) {
    unsigned idx = blockIdx.x * blockDim.x + threadIdx.x;
    int e = idx >> LOG2C;
    int c = idx & (C - 1);
    int s = src[e];
    int d = dst[e];
    float nrm = dinv[s] * dinv[d];
    float v = h[(size_t)s * C + c] * nrm;
    atomAddF(&agg[(size_t)d * C + c], v);
}

// ---------------- self-loop + bias (+ optional ReLU) ----------------
// out may alias agg (same index), so no restrict on those.

template <int C, bool RELU>
__global__ void finish_kernel(const float* agg,
                              const float* __restrict__ h,
                              const float* __restrict__ dinv,
                              const float* __restrict__ bias,
                              float* out) {
    unsigned idx = blockIdx.x * blockDim.x + threadIdx.x;
    int n = idx >> (C == 64 ? 6 : 5);
    int c = idx & (C - 1);
    float di = dinv[n];
    float v = agg[idx] + h[idx] * di * di + bias[c];
    out[idx] = RELU ? fmaxf(v, 0.0f) : v;
}

// ---------------- launcher ----------------

extern "C" void kernel_launch(void* const* d_in, const int* in_sizes, int n_in,
                              void* d_out, int out_size, void* d_ws, size_t ws_size,
                              hipStream_t stream) {
    const float* x    = (const float*)d_in[0];
    const int*   ei   = (const int*)d_in[1];
    const float* W1   = (const float*)d_in[2];
    const float* b1   = (const float*)d_in[3];
    const float* W2   = (const float*)d_in[4];
    const float* b2   = (const float*)d_in[5];
    float*       out  = (float*)d_out;

    const int* srcv = ei;            // edge_index[0]
    const int* dstv = ei + N_EDGES;  // edge_index[1]

    // Workspace layout (floats):
    //   dinv : 100352            (N rounded up for alignment)
    //   h1   : N*64              (layer-1 pre-agg features; later reused: h2 + agg2)
    //   agg1 : N*64              (layer-1 aggregation; holds relu(h1') after finish1)
    float* dinv = (float*)d_ws;
    float* h1   = dinv + 100352;
    float* agg1 = h1 + (size_t)N_NODES * HID_CH;
    float* h2   = h1;                                // reuse (h1 dead after finish1)
    float* agg2 = h1 + (size_t)N_NODES * OUT_CH;     // second half of h1 region

    // 1) degrees -> dinv
    deg_init_kernel<<<(N_NODES + 255) / 256, 256, 0, stream>>>(dinv, N_NODES);
    deg_count_kernel<<<(N_EDGES + 255) / 256, 256, 0, stream>>>(dstv, dinv, N_EDGES);
    rsqrt_kernel<<<(N_NODES + 255) / 256, 256, 0, stream>>>(dinv, N_NODES);

    // 2) h1 = x @ W1   (100000x128 * 128x64), 4 waves/block
    gemm_wmma_f32<IN_CH, HID_CH><<<N_NODES / 16, 128, 0, stream>>>(x, W1, h1);

    // 3) agg1 = scatter_add(h1[src] * norm)
    hipMemsetAsync(agg1, 0, (size_t)N_NODES * HID_CH * sizeof(float), stream);
    edge_agg_kernel<HID_CH, 6>
        <<<(size_t)N_EDGES * HID_CH / 256, 256, 0, stream>>>(h1, srcv, dstv, dinv, agg1);

    // 4) agg1 = relu(agg1 + h1*dinv^2 + b1)   (in place)
    finish_kernel<HID_CH, true>
        <<<(size_t)N_NODES * HID_CH / 256, 256, 0, stream>>>(agg1, h1, dinv, b1, agg1);

    // 5) h2 = agg1 @ W2   (100000x64 * 64x32), 2 waves/block
    gemm_wmma_f32<HID_CH, OUT_CH><<<N_NODES / 16, 64, 0, stream>>>(agg1, W2, h2);

    // 6) agg2 = scatter_add(h2[src] * norm)
    hipMemsetAsync(agg2, 0, (size_t)N_NODES * OUT_CH * sizeof(float), stream);
    edge_agg_kernel<OUT_CH, 5>
        <<<(size_t)N_EDGES * OUT_CH / 256, 256, 0, stream>>>(h2, srcv, dstv, dinv, agg2);

    // 7) out = agg2 + h2*dinv^2 + b2
    finish_kernel<OUT_CH, false>
        <<<(size_t)N_NODES * OUT_CH / 256, 256, 0, stream>>>(agg2, h2, dinv, b2, out);
}